// GCN_50663434224370
// MI455X (gfx1250) — compile-verified
//
#include <hip/hip_runtime.h>
#include <hip/hip_bf16.h>

typedef __attribute__((ext_vector_type(2))) float v2f;
typedef __attribute__((ext_vector_type(8))) float v8f;

// ---------------------------------------------------------------------------
// Degree / normalization kernels
// ---------------------------------------------------------------------------
__global__ void deg_init_kernel(float* __restrict__ dinv, int n) {
    int i = blockIdx.x * blockDim.x + threadIdx.x;
    if (i < n) dinv[i] = 1.0f;   // self-loop contributes 1 to in-degree
}

__global__ void deg_acc_kernel(const int* __restrict__ dst, float* __restrict__ dinv, int e) {
    int i = blockIdx.x * blockDim.x + threadIdx.x;
    if (i < e) atomicAdd(&dinv[dst[i]], 1.0f);
}

__global__ void deg_fin_kernel(float* __restrict__ dinv, int n) {
    int i = blockIdx.x * blockDim.x + threadIdx.x;
    if (i < n) dinv[i] = rsqrtf(dinv[i]);   // deg >= 1 always (self-loop)
}

// ---------------------------------------------------------------------------
// WMMA GEMM:  C[M x (NT*16)] = A[M x K] * B[K x (NT*16)]   (all fp32)
// One wave computes a 16-row tile across all NT column tiles.
// Uses V_WMMA_F32_16X16X4_F32; K must be a multiple of 4.
// A 16x4 layout:  lanes 0-15 -> M=0..15, VGPR0/1 = K={kh, kh+1}, kh = (lane>=16)?2:0
// B 4x16 layout:  row striped across lanes; VGPR0/1 = K={kh, kh+1}
// C/D 16x16:      VGPR i: lanes 0-15 -> M=i, lanes 16-31 -> M=i+8
//
// Tail handling: row m of A only contributes to row m of D, and D rows >= M
// are never stored, so out-of-range lanes just read a clamped (row M-1) address;
// no value masking needed -> unconditional b64 A loads, EXEC stays full.
// ---------------------------------------------------------------------------
template <int NT>
__global__ __launch_bounds__(128) void gemm_wmma_kernel(const float* __restrict__ A,
                                                        const float* __restrict__ B,
                                                        float* __restrict__ C,
                                                        int M, int K) {
    constexpr int NC = NT * 16;
    const int lane = threadIdx.x & 31;
    const int wave = threadIdx.x >> 5;
    const int tile_m = (blockIdx.x * 4 + wave) * 16;
    if (tile_m >= M) return;                       // wave-uniform: EXEC stays full

    const int l15   = lane & 15;
    const int khalf = (lane >> 4) << 1;            // 0 or 2
    int mrow = tile_m + l15;
    if (mrow >= M) mrow = M - 1;                   // clamp address only; data ignored
    const float* arow = A + (size_t)mrow * K + khalf;

    v8f acc[NT];
#pragma unroll
    for (int t = 0; t < NT; ++t) acc[t] = (v8f){0,0,0,0,0,0,0,0};

    for (int k0 = 0; k0 < K; k0 += 4) {
        const float2 av = *reinterpret_cast<const float2*>(arow + k0);  // one b64 load
        v2f a;  a.x = av.x;  a.y = av.y;
        const float* bptr = B + (size_t)(k0 + khalf) * NC + l15;
#pragma unroll
        for (int t = 0; t < NT; ++t) {
            v2f b;
            b.x = bptr[t * 16];
            b.y = bptr[t * 16 + NC];
            acc[t] = __builtin_amdgcn_wmma_f32_16x16x4_f32(
                false, a, false, b, (short)0, acc[t], false, false);
        }
    }

    const int mbase = tile_m + ((lane >> 4) << 3);  // +0 or +8
#pragma unroll
    for (int t = 0; t < NT; ++t) {
#pragma unroll
        for (int i = 0; i < 8; ++i) {
            const int r = mbase + i;
            if (r < M) C[(size_t)r * NC + t * 16 + l15] = acc[t][i];
        }
    }
}

// ---------------------------------------------------------------------------
// agg[i,f] = h[i,f] * dinv[i]^2 + bias[f]     (self-loop term + bias, zero-init)
// ---------------------------------------------------------------------------
template <int F>
__global__ void selfloop_init_kernel(const float* __restrict__ h,
                                     const float* __restrict__ dinv,
                                     const float* __restrict__ bias,
                                     float* __restrict__ agg, int n) {
    constexpr int Q = F / 4;
    long tid = (long)blockIdx.x * blockDim.x + threadIdx.x;
    if (tid >= (long)n * Q) return;
    const int node = (int)(tid / Q);
    const int f4   = (int)(tid % Q) * 4;
    const float s  = dinv[node];
    const float ss = s * s;
    const float4 hv = *reinterpret_cast<const float4*>(h + (size_t)node * F + f4);
    const float4 bv = *reinterpret_cast<const float4*>(bias + f4);
    float4 o;
    o.x = hv.x * ss + bv.x;  o.y = hv.y * ss + bv.y;
    o.z = hv.z * ss + bv.z;  o.w = hv.w * ss + bv.w;
    *reinterpret_cast<float4*>(agg + (size_t)node * F + f4) = o;
}

// ---------------------------------------------------------------------------
// Edge scatter:  agg[dst,f] += h[src,f] * dinv[src]*dinv[dst]
// One thread handles one edge x 4 features (float4 gather, 4 f32 atomics via L2).
// ---------------------------------------------------------------------------
template <int F>
__global__ void edge_agg_kernel(const int* __restrict__ src,
                                const int* __restrict__ dst,
                                const float* __restrict__ dinv,
                                const float* __restrict__ h,
                                float* __restrict__ agg, int e) {
    constexpr int Q = F / 4;
    long tid = (long)blockIdx.x * blockDim.x + threadIdx.x;
    if (tid >= (long)e * Q) return;
    const int ei = (int)(tid / Q);
    const int f4 = (int)(tid % Q) * 4;
    const int s  = src[ei];
    const int d  = dst[ei];
    const float norm = dinv[s] * dinv[d];
    const float4 hv = *reinterpret_cast<const float4*>(h + (size_t)s * F + f4);
    float* o = agg + (size_t)d * F + f4;
    atomicAdd(o + 0, hv.x * norm);
    atomicAdd(o + 1, hv.y * norm);
    atomicAdd(o + 2, hv.z * norm);
    atomicAdd(o + 3, hv.w * norm);
}

// ---------------------------------------------------------------------------
// ReLU (elementwise, float4)
// ---------------------------------------------------------------------------
__global__ void relu_kernel(const float* __restrict__ in, float* __restrict__ out, long n4) {
    long i = (long)blockIdx.x * blockDim.x + threadIdx.x;
    if (i >= n4) return;
    float4 v = reinterpret_cast<const float4*>(in)[i];
    v.x = fmaxf(v.x, 0.0f); v.y = fmaxf(v.y, 0.0f);
    v.z = fmaxf(v.z, 0.0f); v.w = fmaxf(v.w, 0.0f);
    reinterpret_cast<float4*>(out)[i] = v;
}

// ---------------------------------------------------------------------------
extern "C" void kernel_launch(void* const* d_in, const int* in_sizes, int n_in,
                              void* d_out, int out_size, void* d_ws, size_t ws_size,
                              hipStream_t stream) {
    const float* x    = (const float*)d_in[0];   // [N,128]
    const int*   eidx = (const int*)d_in[1];     // [2,E]
    // d_in[2] = edge_attr (unused by reference)
    const float* W1   = (const float*)d_in[3];   // [128,64]
    const float* b1   = (const float*)d_in[4];   // [64]
    const float* W2   = (const float*)d_in[5];   // [64,32]
    const float* b2   = (const float*)d_in[6];   // [32]

    const int N = in_sizes[0] / 128;
    const int E = in_sizes[1] / 2;
    const int* srcp = eidx;
    const int* dstp = eidx + E;

    // Workspace layout (floats): dinv[N] | hbuf[N*64] | agg[N*64]
    float* dinv = (float*)d_ws;
    float* hbuf = dinv + N;
    float* agg  = hbuf + (size_t)N * 64;
    float* out  = (float*)d_out;                 // [N,32]

    const int T = 256;

    // --- degrees / rsqrt norm ---
    deg_init_kernel<<<(N + T - 1) / T, T, 0, stream>>>(dinv, N);
    deg_acc_kernel <<<(E + T - 1) / T, T, 0, stream>>>(dstp, dinv, E);
    deg_fin_kernel <<<(N + T - 1) / T, T, 0, stream>>>(dinv, N);

    // --- layer 1: h = x @ W1  (M=N, K=128, 64 cols -> NT=4) ---
    {
        int mtiles = (N + 15) / 16;
        int blocks = (mtiles + 3) / 4;
        gemm_wmma_kernel<4><<<blocks, 128, 0, stream>>>(x, W1, hbuf, N, 128);
    }
    {
        long work = (long)N * 16;   // N * (64/4)
        selfloop_init_kernel<64><<<(int)((work + T - 1) / T), T, 0, stream>>>(hbuf, dinv, b1, agg, N);
    }
    {
        long work = (long)E * 16;   // E * (64/4)
        edge_agg_kernel<64><<<(int)((work + T - 1) / T), T, 0, stream>>>(srcp, dstp, dinv, hbuf, agg, E);
    }
    {
        long n4 = (long)N * 16;     // N*64/4
        relu_kernel<<<(int)((n4 + T - 1) / T), T, 0, stream>>>(agg, hbuf, n4);
    }

    // --- layer 2: h2 = relu(h1) @ W2  (M=N, K=64, 32 cols -> NT=2), into agg ---
    {
        int mtiles = (N + 15) / 16;
        int blocks = (mtiles + 3) / 4;
        gemm_wmma_kernel<2><<<blocks, 128, 0, stream>>>(hbuf, W2, agg, N, 64);
    }
    {
        long work = (long)N * 8;    // N * (32/4)
        selfloop_init_kernel<32><<<(int)((work + T - 1) / T), T, 0, stream>>>(agg, dinv, b2, out, N);
    }
    {
        long work = (long)E * 8;    // E * (32/4)
        edge_agg_kernel<32><<<(int)((work + T - 1) / T), T, 0, stream>>>(srcp, dstp, dinv, agg, out, E);
    }
}